// Wav2Vec2Quantizer_58875411694322
// MI455X (gfx1250) — compile-verified
//
#include <hip/hip_runtime.h>
#include <math.h>

typedef float v2f __attribute__((ext_vector_type(2)));
typedef float v8f __attribute__((ext_vector_type(8)));

#define M_TOT  65536   // B*S
#define K_PROJ 512     // H
#define N_PROJ 256     // G*D
#define GROUPS 2
#define VCODES 320
#define DGRP   128

// ---------------------------------------------------------------------------
// Kernel 1: h = hidden @ W_proj + b   ([65536,512] x [512,256] -> [65536,256])
// One wave per 64x16 output tile (4 stacked 16x16 WMMA accumulators sharing
// one B fragment): 1.5 VMEM per v_wmma_f32_16x16x4_f32.
// ---------------------------------------------------------------------------
__global__ __launch_bounds__(256) void proj_wmma(const float* __restrict__ hidden,
                                                 const float* __restrict__ W,
                                                 const float* __restrict__ bias,
                                                 float* __restrict__ h) {
  const int lane  = threadIdx.x & 31;
  const int wave  = threadIdx.x >> 5;
  const int tile  = blockIdx.x * 8 + wave;      // (65536/64)*(256/16) = 16384 tiles
  const int ntile = tile & 15;                  // 16 n-tiles
  const int mquad = tile >> 4;                  // 1024 m-quads (64 rows each)
  const int row   = lane & 15;
  const int hi    = lane >> 4;                  // 0: K=k,k+1   1: K=k+2,k+3

  const int n = ntile * 16 + row;               // B column for this lane

  v8f acc[4];
  const float bn = bias[n];
#pragma unroll
  for (int q = 0; q < 4; ++q)
#pragma unroll
    for (int r = 0; r < 8; ++r) acc[q][r] = bn;

  const float* arow0 = hidden + (size_t)(mquad * 64 + row) * K_PROJ + 2 * hi;
  const float* bcol  = W + (size_t)(2 * hi) * N_PROJ + n;

#pragma unroll 2
  for (int k = 0; k < K_PROJ; k += 4) {
    v2f b;
    b.x = bcol[(size_t)k * N_PROJ];             // W[k+2hi,   n]
    b.y = bcol[(size_t)k * N_PROJ + N_PROJ];    // W[k+2hi+1, n]
#pragma unroll
    for (int q = 0; q < 4; ++q) {
      v2f a = *(const v2f*)(arow0 + (size_t)q * 16 * K_PROJ + k);
      acc[q] = __builtin_amdgcn_wmma_f32_16x16x4_f32(false, a, false, b,
                                                     (short)0, acc[q], false, false);
    }
  }

#pragma unroll
  for (int q = 0; q < 4; ++q)
#pragma unroll
    for (int r = 0; r < 8; ++r) {
      const int mm = mquad * 64 + q * 16 + r + 8 * hi;   // C: VGPR r -> rows r / r+8
      h[(size_t)mm * N_PROJ + ntile * 16 + row] = acc[q][r];
    }
}

// ---------------------------------------------------------------------------
// Kernel 2: per (group, 16-row tile): hc via WMMA, distances, argmin (LDS
// ds_min_u64 on packed ordered-float key), one-hot encodings, codevector
// gather into quantized output, usage counts.
// ---------------------------------------------------------------------------
__global__ __launch_bounds__(320) void dist_kernel(const float* __restrict__ h,
                                                   const float* __restrict__ code,
                                                   float* __restrict__ quantized,
                                                   float* __restrict__ enc,
                                                   float* __restrict__ distances,
                                                   float* __restrict__ counts) {
  __shared__ float c2s[VCODES];
  __shared__ float h2s[16];
  __shared__ unsigned long long best[16];

  const int g     = blockIdx.x & 1;
  const int mtile = blockIdx.x >> 1;
  const int m0    = mtile * 16;
  const int tid   = threadIdx.x;
  const int lane  = tid & 31;
  const int wave  = tid >> 5;                   // 10 waves
  const int row   = lane & 15;
  const int hi    = lane >> 4;

  const float* cg = code + (size_t)g * VCODES * DGRP;

  // ||c_v||^2 for this group's 320 codevectors
  if (tid < VCODES) {
    float s = 0.f;
    const float* cv = cg + (size_t)tid * DGRP;
#pragma unroll 4
    for (int d = 0; d < DGRP; ++d) { float x = cv[d]; s += x * x; }
    c2s[tid] = s;
  }
  // ||h_m||^2 for the 16 rows (this group's 128-dim slice)
  if (tid < 16) {
    float s = 0.f;
    const float* hr = h + (size_t)(m0 + tid) * N_PROJ + g * DGRP;
#pragma unroll 4
    for (int d = 0; d < DGRP; ++d) { float x = hr[d]; s += x * x; }
    h2s[tid] = s;
    best[tid] = 0xFFFFFFFFFFFFFFFFull;
  }
  __syncthreads();

  const float* arow = h + (size_t)(m0 + row) * N_PROJ + g * DGRP + 2 * hi;

  // Each wave computes 2 of the 20 N-tiles (320/16).
#pragma unroll
  for (int t = 0; t < 2; ++t) {
    const int nt = wave * 2 + t;
    const int v  = nt * 16 + row;               // this lane's codevector column
    const float* brow = cg + (size_t)v * DGRP + 2 * hi;

    v8f acc = {};
#pragma unroll 8
    for (int k = 0; k < DGRP; k += 4) {
      v2f a = *(const v2f*)(arow + k);
      v2f b = *(const v2f*)(brow + k);          // B[k,n] = code[g][n][k]
      acc = __builtin_amdgcn_wmma_f32_16x16x4_f32(false, a, false, b,
                                                  (short)0, acc, false, false);
    }

#pragma unroll
    for (int r = 0; r < 8; ++r) {
      const int rr = r + 8 * hi;
      const int mm = m0 + rr;
      const float dv = h2s[rr] - 2.0f * acc[r] + c2s[v];
      distances[((size_t)g * M_TOT + mm) * VCODES + v] = dv;
      // order-preserving float -> uint key; index in low bits => argmin
      // tie-breaks to smallest v, matching jnp.argmin.
      unsigned key = __float_as_uint(dv);
      key = (key & 0x80000000u) ? ~key : (key | 0x80000000u);
      const unsigned long long pk = ((unsigned long long)key << 32) | (unsigned)v;
      atomicMin(&best[rr], pk);                 // ds_min_u64 in LDS
    }
  }
  __syncthreads();

  // one-hot encodings
#pragma unroll
  for (int t = 0; t < 2; ++t) {
    const int nt = wave * 2 + t;
    const int v  = nt * 16 + row;
#pragma unroll
    for (int r = 0; r < 8; ++r) {
      const int rr = r + 8 * hi;
      const int mm = m0 + rr;
      const unsigned sel = (unsigned)(best[rr] & 0xFFFFFFFFull);
      enc[((size_t)g * M_TOT + mm) * VCODES + v] = (v == (int)sel) ? 1.0f : 0.0f;
    }
  }

  // gather selected codevectors: 16 rows x 128 dims
  for (int i = tid; i < 16 * DGRP; i += 320) {
    const int rr = i >> 7;
    const int dd = i & (DGRP - 1);
    const unsigned sel = (unsigned)(best[rr] & 0xFFFFFFFFull);
    quantized[(size_t)(m0 + rr) * N_PROJ + g * DGRP + dd] = cg[(size_t)sel * DGRP + dd];
  }

  // codebook usage histogram
  if (tid < 16) {
    const unsigned sel = (unsigned)(best[tid] & 0xFFFFFFFFull);
    atomicAdd(&counts[g * VCODES + sel], 1.0f);
  }
}

// ---------------------------------------------------------------------------
// Kernel 3: perplexity = mean_g exp(-sum_v p log(p+1e-10)), p = clip(counts/BS)
// ---------------------------------------------------------------------------
__global__ __launch_bounds__(640) void perplexity_kernel(const float* __restrict__ counts,
                                                         float* __restrict__ out) {
  __shared__ float gsum[GROUPS];
  const int tid = threadIdx.x;
  if (tid < GROUPS) gsum[tid] = 0.f;
  __syncthreads();
  if (tid < GROUPS * VCODES) {
    const int g = tid / VCODES;
    float p = counts[tid] * (1.0f / (float)M_TOT);
    p = fminf(fmaxf(p, 1e-10f), 1.0f);
    atomicAdd(&gsum[g], p * logf(p + 1e-10f));
  }
  __syncthreads();
  if (tid == 0) {
    float perp = 0.f;
    for (int g = 0; g < GROUPS; ++g) perp += expf(-gsum[g]);
    out[0] = perp * (1.0f / GROUPS);
  }
}

__global__ void zero_counts(float* __restrict__ counts) {
  const int tid = blockIdx.x * blockDim.x + threadIdx.x;
  if (tid < GROUPS * VCODES) counts[tid] = 0.f;
}

// ---------------------------------------------------------------------------
extern "C" void kernel_launch(void* const* d_in, const int* in_sizes, int n_in,
                              void* d_out, int out_size, void* d_ws, size_t ws_size,
                              hipStream_t stream) {
  const float* hidden = (const float*)d_in[0];   // [16,4096,512]
  const float* W      = (const float*)d_in[1];   // [512,256]
  const float* bias   = (const float*)d_in[2];   // [256]
  const float* code   = (const float*)d_in[3];   // [2,320,128]

  float* out       = (float*)d_out;
  float* quantized = out;                                            // 16,777,216
  float* enc       = out + (size_t)16777216;                         // 41,943,040
  float* distances = out + (size_t)16777216 + 41943040;              // 41,943,040
  float* perp      = out + (size_t)16777216 + 41943040 + 41943040;   // 1

  float* h      = (float*)d_ws;                       // 65536*256 f32 = 64 MB
  float* counts = h + (size_t)M_TOT * N_PROJ;         // 640 f32

  zero_counts<<<1, 640, 0, stream>>>(counts);
  proj_wmma<<<(M_TOT / 64) * (N_PROJ / 16) / 8, 256, 0, stream>>>(hidden, W, bias, h);
  dist_kernel<<<GROUPS * (M_TOT / 16), 320, 0, stream>>>(h, code, quantized, enc,
                                                         distances, counts);
  perplexity_kernel<<<1, 640, 0, stream>>>(counts, perp);
}